// SubTokenEmbedding_2370821947990
// MI455X (gfx1250) — compile-verified
//
#include <hip/hip_runtime.h>

// SubTokenEmbedding: out[n, :] = sum_l table[subtokens[n,l], :], padding row 0 -> 0.
// Memory-bound gather: table (12.8 MB) is L2-resident; HBM ~128 MB => ~5.5us floor.
// CDNA5 paths used: global_load_async_to_lds_b128 (+ s_wait_asynccnt) for index
// staging, NT store for the 102 MB output stream so it doesn't evict the table.

#define PADDING_INDEX 0
#define LSUB 8            // MAX_SUBTOKENS
#define DIM 64            // OUT_SIZE (16 float4 per row)
#define NODES_PER_BLOCK 16
#define THREADS 256       // 8 wave32s; 16 lanes per node

typedef float v4f __attribute__((ext_vector_type(4)));
typedef int   v4i __attribute__((ext_vector_type(4)));

__global__ __launch_bounds__(THREADS)
void subtoken_embed_sum_kernel(const int* __restrict__ subtokens,   // [N, 8]
                               const float* __restrict__ table,     // [VOCAB, 64]
                               float* __restrict__ out,             // [N, 64]
                               int num_nodes)
{
    __shared__ __align__(16) int s_idx[NODES_PER_BLOCK * LSUB];     // 512 B

    const int tid         = threadIdx.x;
    const int block_node0 = blockIdx.x * NODES_PER_BLOCK;

    // ---- Stage this block's 128 indices (512 B) into LDS ----
    const bool full_block = (block_node0 + NODES_PER_BLOCK) <= num_nodes;
    if (full_block) {
        if (tid < 32) {
            // One wave-wide async copy: 32 lanes x 16 B = 512 B, tracked on ASYNCcnt.
            const int* gsrc = subtokens + (size_t)block_node0 * LSUB + tid * 4;
            // Low 32 bits of the generic pointer to LDS == LDS byte offset.
            unsigned lds_off = (unsigned)(unsigned long long)(&s_idx[tid * 4]);
            asm volatile("global_load_async_to_lds_b128 %0, %1, off"
                         :
                         : "v"(lds_off), "v"(gsrc)
                         : "memory");
            asm volatile("s_wait_asynccnt 0x0" ::: "memory");
        }
    } else {
        // Tail block: guarded scalar staging; missing slots get the padding index
        // (which contributes exactly zero, matching the reference).
        if (tid < NODES_PER_BLOCK * LSUB) {
            long long g = (long long)block_node0 * LSUB + tid;
            s_idx[tid] = (g < (long long)num_nodes * LSUB) ? subtokens[g]
                                                           : PADDING_INDEX;
        }
    }
    __syncthreads();

    const int node  = block_node0 + (tid >> 4);
    const int chunk = tid & 15;                 // which float4 of the 16 in a row
    if (node >= num_nodes) return;

    // 8 indices for this node, read from LDS as two b128 loads (broadcast across
    // the 16 lanes of the node).
    const v4i* sidx4 = (const v4i*)&s_idx[(tid >> 4) * LSUB];
    v4i ia = sidx4[0];
    v4i ib = sidx4[1];
    int idx[LSUB] = {ia.x, ia.y, ia.z, ia.w, ib.x, ib.y, ib.z, ib.w};

    const v4f* tab4 = (const v4f*)table;

    v4f acc = (v4f)(0.0f);
#pragma unroll
    for (int l = 0; l < LSUB; ++l) {
        // Always load (row 0 is valid memory), mask branchlessly: no divergence,
        // all 8 global_load_b128 can be issued back-to-back to hide L2 latency.
        v4f v  = tab4[(size_t)idx[l] * (DIM / 4) + chunk];
        float m = (idx[l] != PADDING_INDEX) ? 1.0f : 0.0f;
        acc.x = fmaf(m, v.x, acc.x);
        acc.y = fmaf(m, v.y, acc.y);
        acc.z = fmaf(m, v.z, acc.z);
        acc.w = fmaf(m, v.w, acc.w);
    }

    // NT store: output is write-once / never re-read; keep the 12.8 MB table
    // resident in L2 instead of letting the 102 MB output stream thrash it.
    v4f* out4 = (v4f*)out;
    __builtin_nontemporal_store(acc, &out4[(size_t)node * (DIM / 4) + chunk]);
}

extern "C" void kernel_launch(void* const* d_in, const int* in_sizes, int n_in,
                              void* d_out, int out_size, void* d_ws, size_t ws_size,
                              hipStream_t stream) {
    const int*   subtokens = (const int*)d_in[0];     // [N, 8] int32
    const float* table     = (const float*)d_in[1];   // [VOCAB, 64] f32
    float*       out       = (float*)d_out;           // [N, 64] f32

    const int num_nodes = in_sizes[0] / LSUB;
    const int grid      = (num_nodes + NODES_PER_BLOCK - 1) / NODES_PER_BLOCK;

    subtoken_embed_sum_kernel<<<grid, THREADS, 0, stream>>>(subtokens, table, out,
                                                            num_nodes);
}